// WalkingManController_88965952569385
// MI455X (gfx1250) — compile-verified
//
#include <hip/hip_runtime.h>

// CDNA5 (gfx1250) WMMA types
typedef float v2f __attribute__((ext_vector_type(2)));
typedef float v8f __attribute__((ext_vector_type(8)));

#define N_AGENTS 8192
#define K_COMP   4096
#define ROWS_PER_BLOCK 16
#define THREADS 128   // 4 wave32s

// Problem constants (from reference)
constexpr float Hf     = 0.05f;
constexpr float SIG    = 5.0f / 64.0f;               // sigma = L/RES
constexpr float LOG2E  = 1.44269504088896340736f;
constexpr float TWO_PI = 6.28318530717958647692f;
// exp(-d^2/(2h^2)) = exp2(-d^2 * S), S folded with log2(e): one v_exp_f32 per element
constexpr float SH    = LOG2E / (2.0f * Hf * Hf);
constexpr float SS    = LOG2E / (2.0f * SIG * SIG);
constexpr float CW    = 1.0f / (TWO_PI * SIG * SIG);                    // gmm component norm
constexpr float CK    = 1.0f / ((float)N_AGENTS * TWO_PI * Hf * Hf);   // kde norm (1/(N*2pi*h^2))
constexpr float CKH   = CK / (Hf * Hf);
constexpr float INVS2 = 1.0f / (SIG * SIG);

__global__ __launch_bounds__(THREADS)
void wmc_kernel(const float* __restrict__ t,
                const float* __restrict__ X,
                const float* __restrict__ m0,
                const float* __restrict__ mv,
                const float* __restrict__ w,
                float* __restrict__ out)
{
    // 64KB LDS: Xx at [0,8192), Xy at [8192,16384). Front 512 floats reused for
    // the cross-wave reduction after all sweeps complete (barrier-protected).
    __shared__ float lds[16384];

    const int tid    = threadIdx.x;
    const int lane   = tid & 31;
    const int wave   = tid >> 5;
    const int low16  = (lane < 16);
    const int colsel = lane & 15;
    const int i0     = blockIdx.x * ROWS_PER_BLOCK;

    // ---- Stage X (SoA) into LDS: 8192 agents / 128 threads, uniform trip count ----
    for (int k = tid; k < N_AGENTS; k += THREADS) {
        float2 xv = ((const float2*)X)[k];
        lds[k]          = xv.x;
        lds[8192 + k]   = xv.y;
    }

    const float t0 = t[0];

    // ---- A fragments (16x4 f32 layout: VGPR0 = K0/K2, VGPR1 = K1/K3 per half-wave) ----
    // Row i: [2s*x_i, 2s*y_i, -s, -s*||X_i||^2]  so  (A.B)_ik = -s*d_ik^2 (log2 domain)
    float2 xa  = ((const float2*)X)[i0 + colsel];
    float  ar2 = xa.x * xa.x + xa.y * xa.y;
    v2f aK, aG;
    aK.x = low16 ? (2.0f * SH) * xa.x : -SH;
    aK.y = low16 ? (2.0f * SH) * xa.y : -SH * ar2;
    aG.x = low16 ? (2.0f * SS) * xa.x : -SS;
    aG.y = low16 ? (2.0f * SS) * xa.y : -SS * ar2;

    // Lane-local accumulators: acc[sum][C-slot]
    // 0:Σe  1:Σe·x  2:Σe·y  3:Σc  4:Σc·mx  5:Σc·my  6:Σc·mvx  7:Σc·mvy
    float acc[8][8];
    #pragma unroll
    for (int a = 0; a < 8; ++a) {
        #pragma unroll
        for (int v = 0; v < 8; ++v) acc[a][v] = 0.0f;
    }

    __syncthreads();

    // ---- KDE sweep: 512 column tiles of 16, 128 per wave ----
    for (int jt = wave * 128; jt < wave * 128 + 128; ++jt) {
        int   col = jt * 16 + colsel;
        float cx  = lds[col];
        float cy  = lds[8192 + col];
        float cr2 = cx * cx + cy * cy;
        // B col k: [x, y, ||col||^2, 1]  (4x16 f32: VGPR0 = K0/K2, VGPR1 = K1/K3)
        v2f b;
        b.x = low16 ? cx : cr2;
        b.y = low16 ? cy : 1.0f;
        v8f q = {};
        q = __builtin_amdgcn_wmma_f32_16x16x4_f32(false, aK, false, b,
                                                  (short)0, q, false, false);
        #pragma unroll
        for (int v = 0; v < 8; ++v) {
            float e = __builtin_amdgcn_exp2f(q[v]);   // exp(-d^2/2h^2)
            acc[0][v] += e;
            acc[1][v]  = fmaf(e, cx, acc[1][v]);
            acc[2][v]  = fmaf(e, cy, acc[2][v]);
        }
    }

    // ---- GMM sweep: 256 column tiles of 16, 64 per wave (m = m0 + mv*t on the fly) ----
    for (int jt = wave * 64; jt < wave * 64 + 64; ++jt) {
        int    col = jt * 16 + colsel;
        float2 m0v = ((const float2*)m0)[col];
        float2 mvv = ((const float2*)mv)[col];
        float  wk  = w[col] * CW;
        float  mx  = fmaf(mvv.x, t0, m0v.x);
        float  my  = fmaf(mvv.y, t0, m0v.y);
        float  mr2 = mx * mx + my * my;
        v2f b;
        b.x = low16 ? mx : mr2;
        b.y = low16 ? my : 1.0f;
        v8f q = {};
        q = __builtin_amdgcn_wmma_f32_16x16x4_f32(false, aG, false, b,
                                                  (short)0, q, false, false);
        #pragma unroll
        for (int v = 0; v < 8; ++v) {
            float c = wk * __builtin_amdgcn_exp2f(q[v]);  // w_k/(2πσ²)·exp(-d²/2σ²)
            acc[3][v] += c;
            acc[4][v]  = fmaf(c, mx,    acc[4][v]);
            acc[5][v]  = fmaf(c, my,    acc[5][v]);
            acc[6][v]  = fmaf(c, mvv.x, acc[6][v]);
            acc[7][v]  = fmaf(c, mvv.y, acc[7][v]);
        }
    }

    // ---- Reduce across the 16 lanes of each half-wave (wave32 xor-shuffle) ----
    #pragma unroll
    for (int a = 0; a < 8; ++a) {
        #pragma unroll
        for (int v = 0; v < 8; ++v) {
            float s = acc[a][v];
            s += __shfl_xor(s, 1, 32);
            s += __shfl_xor(s, 2, 32);
            s += __shfl_xor(s, 4, 32);
            s += __shfl_xor(s, 8, 32);
            acc[a][v] = s;
        }
    }

    __syncthreads();   // all waves past both sweeps; safe to reuse lds[0..511]

    // C layout: slot v, lanes 0-15 -> row v ; lanes 16-31 -> row v+8.
    // One lane per half-wave publishes all 8 slots x 8 sums.
    if (colsel == 0) {
        int half8 = (lane >> 4) * 8;
        #pragma unroll
        for (int a = 0; a < 8; ++a) {
            #pragma unroll
            for (int v = 0; v < 8; ++v)
                lds[wave * 128 + a * 16 + half8 + v] = acc[a][v];
        }
    }
    __syncthreads();

    // ---- Final analytic combine, one thread per row ----
    if (tid < ROWS_PER_BLOCK) {
        int row = tid;
        float S[8];
        #pragma unroll
        for (int a = 0; a < 8; ++a)
            S[a] = lds[0 * 128 + a * 16 + row] + lds[1 * 128 + a * 16 + row]
                 + lds[2 * 128 + a * 16 + row] + lds[3 * 128 + a * 16 + row];

        float2 xi = ((const float2*)X)[i0 + row];
        float fh  = CK * S[0];                          // f_hat (self term e=1 included)
        // grad_kde = (CK/h^2) * (Σe·Xn − X_i·Σe)
        float gkx = CKH * (S[1] - xi.x * S[0]);
        float gky = CKH * (S[2] - xi.y * S[0]);
        // grad of (−gmm_density) = (1/σ²) * (X_i·Σc − Σc·m)
        float ggx = INVS2 * (xi.x * S[3] - S[4]);
        float ggy = INVS2 * (xi.y * S[3] - S[5]);
        // out = (feed_back + feed_forward)/f_hat ; feed_back = −phi_grad, ff = Σc·mv
        float ox = (-(gkx + ggx) + S[6]) / fh;
        float oy = (-(gky + ggy) + S[7]) / fh;
        out[2 * (i0 + row) + 0] = ox;
        out[2 * (i0 + row) + 1] = oy;
    }
}

extern "C" void kernel_launch(void* const* d_in, const int* in_sizes, int n_in,
                              void* d_out, int out_size, void* d_ws, size_t ws_size,
                              hipStream_t stream)
{
    const float* t  = (const float*)d_in[0];
    const float* X  = (const float*)d_in[1];
    const float* m0 = (const float*)d_in[2];
    const float* mv = (const float*)d_in[3];
    const float* w  = (const float*)d_in[4];
    float* out = (float*)d_out;

    wmc_kernel<<<N_AGENTS / ROWS_PER_BLOCK, THREADS, 0, stream>>>(t, X, m0, mv, w, out);
}